// StackGCNs_23115513987505
// MI455X (gfx1250) — compile-verified
//
#include <hip/hip_runtime.h>
#include <cstdint>
#include <cstddef>

// resSumGCN: h_{i+1} = A h_i (sparse scatter SpMM), out = h1 + h2 + h3.
// Inputs: d_in[0]=x (N*64 f32), d_in[1]=edge_row (E i32), d_in[2]=edge_col (E i32),
//         d_in[3]=edge_vals (E f32). Output: N*64 f32.
//
// Strategy (MI455X): x/h (25.6 MB) are L2-resident (192 MB L2), so random
// gathers stay on-chip. Per edge: 16 lanes (half wave32) each gather a float4
// slice of the 64 features and scatter-add with native f32 atomics. Edge
// metadata is streamed through LDS with CDNA5 async global->LDS B128 copies
// (ASYNCcnt + s_wait_asynccnt).

#define EDGES_PER_BLOCK 1024
#define THREADS 256

__global__ __launch_bounds__(THREADS) void spmm_scatter_kernel(
    const int* __restrict__ erow,
    const int* __restrict__ ecol,
    const float* __restrict__ evals,
    const float* __restrict__ hin,
    float* __restrict__ hout,
    int E)
{
    // 3 x 4KB staging buffers: rows, cols, vals for 1024 edges
    __shared__ __align__(16) unsigned smem[3 * EDGES_PER_BLOCK];

    const int tid   = threadIdx.x;
    const int chunk = blockIdx.x * EDGES_PER_BLOCK;

    // ---- async stage: each thread copies 4 edges (16B) of each array ----
    int gi = chunk + tid * 4;
    int clampv = E - 4;               // E is a multiple of 4 (1.6M)
    if (gi > clampv) gi = clampv;     // tail blocks: clamp (guarded at use)

    unsigned lbase = (unsigned)(unsigned long long)(uintptr_t)smem; // LDS byte offset
    unsigned ldst  = lbase + (unsigned)tid * 16u;
    unsigned long long a_row = (unsigned long long)(uintptr_t)(erow  + gi);
    unsigned long long a_col = (unsigned long long)(uintptr_t)(ecol  + gi);
    unsigned long long a_val = (unsigned long long)(uintptr_t)(evals + gi);

    asm volatile("global_load_async_to_lds_b128 %0, %1, off"
                 :: "v"(ldst),          "v"(a_row) : "memory");
    asm volatile("global_load_async_to_lds_b128 %0, %1, off"
                 :: "v"(ldst + 4096u),  "v"(a_col) : "memory");
    asm volatile("global_load_async_to_lds_b128 %0, %1, off"
                 :: "v"(ldst + 8192u),  "v"(a_val) : "memory");
    asm volatile("s_wait_asynccnt 0" ::: "memory");
    __syncthreads();

    const unsigned* s_row = smem;
    const unsigned* s_col = smem + EDGES_PER_BLOCK;
    const float*    s_val = (const float*)(smem + 2 * EDGES_PER_BLOCK);

    const int part = tid & 15;   // which float4 slice of the 64 features
    const int esub = tid >> 4;   // 0..15: edge sub-slot within a pass
    const float4* __restrict__ hin4 = (const float4*)hin;

    #pragma unroll 4
    for (int k = 0; k < EDGES_PER_BLOCK / 16; ++k) {
        const int el = (k << 4) + esub;           // 0..1023
        if (chunk + el >= E) break;               // tail guard

        const unsigned row = s_row[el];           // LDS broadcast to 16 lanes
        const unsigned col = s_col[el];
        const float    v   = s_val[el];

        float4 f = hin4[(size_t)col * 16 + part]; // coalesced 256B/edge gather (L2)
        float* dst = hout + (size_t)row * 64 + (size_t)part * 4;

        unsafeAtomicAdd(dst + 0, f.x * v);        // native global_atomic_add_f32
        unsafeAtomicAdd(dst + 1, f.y * v);
        unsafeAtomicAdd(dst + 2, f.z * v);
        unsafeAtomicAdd(dst + 3, f.w * v);
    }
}

__global__ __launch_bounds__(256) void accum_add_kernel(
    const float4* __restrict__ h, float4* __restrict__ out, int n4)
{
    int i = blockIdx.x * blockDim.x + threadIdx.x;
    if (i < n4) {
        float4 a = out[i];
        float4 b = h[i];
        a.x += b.x; a.y += b.y; a.z += b.z; a.w += b.w;
        out[i] = a;
    }
}

extern "C" void kernel_launch(void* const* d_in, const int* in_sizes, int n_in,
                              void* d_out, int out_size, void* d_ws, size_t ws_size,
                              hipStream_t stream)
{
    const float* x     = (const float*)d_in[0];
    const int*   erow  = (const int*)d_in[1];
    const int*   ecol  = (const int*)d_in[2];
    const float* evals = (const float*)d_in[3];
    float* out = (float*)d_out;

    const int E          = in_sizes[1];
    const int feat_total = in_sizes[0];           // N * 64 == out_size
    const size_t bytes   = (size_t)feat_total * sizeof(float);

    float* bufA = (float*)d_ws;                   // h ping
    float* bufB = bufA + feat_total;              // h pong (needs 2*25.6MB of ws)

    const int n4          = feat_total / 4;
    const int spmm_blocks = (E + EDGES_PER_BLOCK - 1) / EDGES_PER_BLOCK;
    const int add_blocks  = (n4 + 255) / 256;

    // out = 0
    hipMemsetAsync(d_out, 0, bytes, stream);

    // h1 = A x ; out += h1
    hipMemsetAsync(bufA, 0, bytes, stream);
    spmm_scatter_kernel<<<spmm_blocks, THREADS, 0, stream>>>(erow, ecol, evals, x, bufA, E);
    accum_add_kernel<<<add_blocks, 256, 0, stream>>>((const float4*)bufA, (float4*)out, n4);

    // h2 = A h1 ; out += h2
    hipMemsetAsync(bufB, 0, bytes, stream);
    spmm_scatter_kernel<<<spmm_blocks, THREADS, 0, stream>>>(erow, ecol, evals, bufA, bufB, E);
    accum_add_kernel<<<add_blocks, 256, 0, stream>>>((const float4*)bufB, (float4*)out, n4);

    // h3 = A h2 ; out += h3
    hipMemsetAsync(bufA, 0, bytes, stream);
    spmm_scatter_kernel<<<spmm_blocks, THREADS, 0, stream>>>(erow, ecol, evals, bufB, bufA, E);
    accum_add_kernel<<<add_blocks, 256, 0, stream>>>((const float4*)bufA, (float4*)out, n4);
}